// BSplineTransform_55645596287738
// MI455X (gfx1250) — compile-verified
//
#include <hip/hip_runtime.h>
#include <stdint.h>

// BSplineTransform (order-1 grid_pull) for MI455X / gfx1250.
// Memory-bound gather: LDS-staged control grid (via TDM async-tensor DMA),
// NT-hinted b128 streaming of coordinates/output, packed-f32 bilerp,
// 4x-unrolled counted main loop for deep VMEM clauses.

typedef float    v2f __attribute__((ext_vector_type(2)));
typedef float    v4f __attribute__((ext_vector_type(4)));
typedef uint32_t v4u __attribute__((ext_vector_type(4)));
typedef int      v4i __attribute__((ext_vector_type(4)));
typedef int      v8i __attribute__((ext_vector_type(8)));

#define HC     64
#define WC     64
#define NCTL   (HC * WC)            // 4096 control points per channel
#define NPIX   (16 * 1024 * 1024)   // B*H*W
#define NVEC   (NPIX / 2)           // float4 groups (2 pixels each)
#define NBLK   1024
#define NTHR   256
#define STRIDE (NBLK * NTHR)        // 262144 threads
#define NITER  (NVEC / STRIDE)      // exactly 32 iterations per thread

// ---------------------------------------------------------------------------
// TDM: 1-D tensor DMA of `nelem` contiguous 4-byte elements -> LDS.
// Descriptor per CDNA5 ISA:
//   group0: [1:0]=count=1, [63:32]=lds_addr, [120:64]=global_addr, [127:126]=type=2
//   group1: data_size=2 (4B), tensor_dim0=nelem, tensor_dim1=1,
//           tile_dim0=nelem, tile_dim1=1, tensor_dim0_stride=nelem
// ---------------------------------------------------------------------------
__device__ __forceinline__ void tdm_load_1d(uint32_t lds_off, const void* gptr,
                                            uint32_t nelem) {
  uint64_t ga = (uint64_t)(uintptr_t)gptr;
  v4u g0;
  g0.x = 1u;                                               // count=1, user mode
  g0.y = lds_off;                                          // LDS byte address
  g0.z = (uint32_t)ga;                                     // global_addr[31:0]
  g0.w = (uint32_t)((ga >> 32) & 0x1FFFFFFu) | (2u << 30); // addr[56:32] | type=2
  v8i g1;
  g1[0] = (int)(2u << 16);                                 // data_size=2 -> 4 bytes
  g1[1] = (int)((nelem & 0xFFFFu) << 16);                  // tensor_dim0[15:0]
  g1[2] = (int)((nelem >> 16) | (1u << 16));               // tensor_dim0[31:16] | tensor_dim1=1
  g1[3] = (int)((nelem & 0xFFFFu) << 16);                  // tile_dim0 = nelem
  g1[4] = 1;                                               // tile_dim1=1, tile_dim2=0
  g1[5] = (int)nelem;                                      // tensor_dim0_stride[31:0]
  g1[6] = (int)((nelem & 0xFFFFu) << 16);                  // tensor_dim1_stride[15:0]
  g1[7] = 0;
  v4i g2 = {0, 0, 0, 0};
  v4i g3 = {0, 0, 0, 0};
#if defined(__clang_major__) && (__clang_major__ >= 23)
  v8i g4 = {0, 0, 0, 0, 0, 0, 0, 0};
  __builtin_amdgcn_tensor_load_to_lds(g0, g1, g2, g3, g4, 0);
#else
  __builtin_amdgcn_tensor_load_to_lds(g0, g1, g2, g3, 0);
#endif
}

// ---------------------------------------------------------------------------
// Bilinear sample of the interleaved {ch0,ch1} control grid held in LDS.
// Factored lerp == reference 4-corner zero-bound form for in-FOV coords
// (any OOB corner has weight exactly 0); whole result masked by in-FOV.
// ---------------------------------------------------------------------------
__device__ __forceinline__ v2f sample_px(float x, float y,
                                         const v2f* __restrict__ cp) {
  float fx = __builtin_floorf(x);
  float fy = __builtin_floorf(y);
  float wx = x - fx;
  float wy = y - fy;
  int ix = (int)fx;
  int iy = (int)fy;
  int ixp = ix + 1, iyp = iy + 1;
  int ix0 = ix  < 0 ? 0 : (ix  > HC - 1 ? HC - 1 : ix );
  int ix1 = ixp < 0 ? 0 : (ixp > HC - 1 ? HC - 1 : ixp);
  int iy0 = iy  < 0 ? 0 : (iy  > WC - 1 ? WC - 1 : iy );
  int iy1 = iyp < 0 ? 0 : (iyp > WC - 1 ? WC - 1 : iyp);
  v2f c00 = cp[ix0 * WC + iy0];   // ds_load_b64 gathers
  v2f c01 = cp[ix0 * WC + iy1];
  v2f c10 = cp[ix1 * WC + iy0];
  v2f c11 = cp[ix1 * WC + iy1];
  v2f a = c00 + wy * (c01 - c00); // packed f32 fma
  v2f b = c10 + wy * (c11 - c10);
  v2f r = a + wx * (b - a);
  float m = (x >= 0.0f && x <= (float)(HC - 1) &&
             y >= 0.0f && y <= (float)(WC - 1)) ? 1.0f : 0.0f;
  return r * m;
}

__global__ __launch_bounds__(NTHR) void
BSplineTransform_55645596287738_kernel(const float* __restrict__ coords,
                                       const float* __restrict__ ctrl,
                                       const float* __restrict__ disp,
                                       float* __restrict__ out) {
  __shared__ float lA[2 * NCTL];  // raw control_points (TDM destination)
  __shared__ float lB[2 * NCTL];  // raw displacements (TDM destination)
  __shared__ float lC[2 * NCTL];  // summed grid, interleaved {ch0,ch1} pairs

  const int t = threadIdx.x;

  // Zero-init the DMA destinations (keeps the LDS reads well-defined for the
  // compiler; the TDM transfer overwrites with real data).
  for (int j = t; j < 2 * NCTL; j += NTHR) {
    lA[j] = 0.0f;
    lB[j] = 0.0f;
  }
  __syncthreads();

  // One wave per workgroup issues the two tensor DMAs and waits on TENSORcnt.
  if (t < 32) {
    uint32_t offA = (uint32_t)(uintptr_t)&lA[0];
    uint32_t offB = (uint32_t)(uintptr_t)&lB[0];
    tdm_load_1d(offA, ctrl, 2 * NCTL);
    tdm_load_1d(offB, disp, 2 * NCTL);
    __builtin_amdgcn_s_wait_tensorcnt(0);
  }
  __syncthreads();

  // cp = control_points + displacements, interleaved so one ds_load_b64
  // fetches both channels of a corner.
  for (int j = t; j < NCTL; j += NTHR) {
    lC[2 * j + 0] = lA[j]        + lB[j];
    lC[2 * j + 1] = lA[NCTL + j] + lB[NCTL + j];
  }
  __syncthreads();

  const v2f* cp   = (const v2f*)lC;
  const v4f* cin  = (const v4f*)coords;
  v4f*       cout = (v4f*)out;

  // Counted loop with compile-time trip count; unroll so the compiler can
  // clause multiple b128 NT loads/stores and keep several VMEM ops in flight
  // per wave (bandwidth-bound at ~23.3 TB/s).
  const int base = (int)(blockIdx.x * NTHR) + t;
#pragma unroll 4
  for (int k = 0; k < NITER; ++k) {
    const int i = base + k * STRIDE;
    v4f c = __builtin_nontemporal_load(cin + i);  // 2 pixels of coordinates
    v2f r0 = sample_px(c.x, c.y, cp);
    v2f r1 = sample_px(c.z, c.w, cp);
    v4f r;
    r.x = r0.x; r.y = r0.y; r.z = r1.x; r.w = r1.y;
    __builtin_nontemporal_store(r, cout + i);     // streaming output
  }
}

extern "C" void kernel_launch(void* const* d_in, const int* in_sizes, int n_in,
                              void* d_out, int out_size, void* d_ws, size_t ws_size,
                              hipStream_t stream) {
  const float* coords = (const float*)d_in[0];  // (16,1024,1024,2) f32
  const float* ctrl   = (const float*)d_in[1];  // (2,64,64) f32
  const float* disp   = (const float*)d_in[2];  // (2,64,64) f32
  float*       out    = (float*)d_out;          // (16,1024,1024,2) f32

  dim3 grid(NBLK), block(NTHR);
  BSplineTransform_55645596287738_kernel<<<grid, block, 0, stream>>>(
      coords, ctrl, disp, out);
}